// StateExtract_7791070675548
// MI455X (gfx1250) — compile-verified
//
#include <hip/hip_runtime.h>
#include <hip/hip_bf16.h>

// ---------------------------------------------------------------------------
// Heterogeneous GATv2 forward for MI455X (gfx1250, wave32).
// Dense GEMMs: V_WMMA_F32_16X16X4_F32, one wave = 16x64 output strip
// (A fragment reused across 4 back-to-back WMMAs per K-step).
// Edge softmax/aggregation: float4 (b128) gathers + L2 float atomics.
// ---------------------------------------------------------------------------

typedef float v2f __attribute__((ext_vector_type(2)));
typedef float v8f __attribute__((ext_vector_type(8)));

#define NEG_GAT 0.2f
#define NEG_LR  0.01f

#define NB    64
#define NL    3
#define NH    64
#define NG    128
#define NGG   256
#define N_OP  100000
#define N_M   10000
#define N_A   10000
#define E_PP  800000
#define E_OM  800000
#define E_MA  400000

// ---- monotone float <-> uint mapping for atomic segment-max ---------------
__device__ __forceinline__ unsigned int f2ord(float f) {
  unsigned int u = __float_as_uint(f);
  return (u & 0x80000000u) ? ~u : (u | 0x80000000u);
}
__device__ __forceinline__ float ord2f(unsigned int k) {
  return (k & 0x80000000u) ? __uint_as_float(k ^ 0x80000000u)
                           : __uint_as_float(~k);
}

// ---------------------------------------------------------------------------
// WMMA fp32 GEMM: Out[N,CO] = A[N,K] @ W[K,CO] + bias, optional leaky-relu.
// One wave computes a 16x64 strip = 4 x (16x16) tiles sharing one A fragment.
// K stepped by 4 via V_WMMA_F32_16X16X4_F32.
// Fragment layout per CDNA5 ISA 7.12.2:
//   A 16x4:  v0 = {K=k0 (lanes 0-15), K=k0+2 (lanes 16-31)}, v1 = {k0+1,k0+3}
//   B 4x16:  same K split, row striped across lanes (N = lane&15)
//   C/D:     VGPR v holds (M=v, lanes 0-15) and (M=v+8, lanes 16-31)
// Requires K % 4 == 0 and CO % 64 == 0 (true for all call sites here).
// ---------------------------------------------------------------------------
__global__ void gemm_wmma_f32(const float* __restrict__ A,
                              const float* __restrict__ W,
                              const float* __restrict__ bias,
                              float* __restrict__ Out,
                              int N, int K, int CO, int act) {
  int wave = threadIdx.x >> 5;
  int lane = threadIdx.x & 31;
  int half = lane >> 4;
  int r    = lane & 15;
  int m0 = (blockIdx.x * 4 + wave) * 16;
  int n0 = blockIdx.y * 64;

  int rowA = m0 + r;
  if (rowA >= N) rowA = N - 1;             // clamp: keeps EXEC all-ones
  const float* arow  = A + (long)rowA * K;
  const float* wbase = W + n0 + r;

  v8f acc0 = {0.f, 0.f, 0.f, 0.f, 0.f, 0.f, 0.f, 0.f};
  v8f acc1 = acc0, acc2 = acc0, acc3 = acc0;

  for (int k0 = 0; k0 < K; k0 += 4) {
    int ka = k0 + 2 * half;
    v2f a;
    a.x = arow[ka];
    a.y = arow[ka + 1];
    const float* w0 = wbase + (long)ka * CO;
    const float* w1 = wbase + (long)(ka + 1) * CO;
    v2f b0; b0.x = w0[0];  b0.y = w1[0];
    v2f b1; b1.x = w0[16]; b1.y = w1[16];
    v2f b2; b2.x = w0[32]; b2.y = w1[32];
    v2f b3; b3.x = w0[48]; b3.y = w1[48];
    acc0 = __builtin_amdgcn_wmma_f32_16x16x4_f32(false, a, false, b0, (short)0, acc0, false, false);
    acc1 = __builtin_amdgcn_wmma_f32_16x16x4_f32(false, a, false, b1, (short)0, acc1, false, false);
    acc2 = __builtin_amdgcn_wmma_f32_16x16x4_f32(false, a, false, b2, (short)0, acc2, false, false);
    acc3 = __builtin_amdgcn_wmma_f32_16x16x4_f32(false, a, false, b3, (short)0, acc3, false, false);
  }

  v8f accs[4] = {acc0, acc1, acc2, acc3};
#pragma unroll
  for (int t = 0; t < 4; ++t) {
    int col = n0 + t * 16 + r;
    float bv = bias ? bias[col] : 0.f;
#pragma unroll
    for (int v = 0; v < 8; ++v) {
      int row = m0 + v + 8 * half;
      if (row < N) {
        float val = accs[t][v] + bv;
        if (act) val = val > 0.f ? val : NEG_LR * val;
        Out[(long)row * CO + col] = val;
      }
    }
  }
}

// ---------------------------------------------------------------------------
// Edge kernels (GATv2 attention). src == nullptr means identity (edge e -> e).
// xr_stride == 0 broadcasts a single xr row (graph-token readout).
// All rows are 16B aligned (C % 4 == 0, 256B-aligned buffers) -> b128 loads.
// ---------------------------------------------------------------------------
__global__ void edge_score_kernel(const float* __restrict__ xl,
                                  const float* __restrict__ xr, int xr_stride,
                                  const int* __restrict__ src,
                                  const int* __restrict__ dst,
                                  const float* __restrict__ ea,
                                  const float* __restrict__ We, int edge_dim,
                                  const float* __restrict__ att,
                                  float* __restrict__ score,
                                  unsigned int* __restrict__ maxbuf,
                                  int E, int C) {
  int e = blockIdx.x * blockDim.x + threadIdx.x;
  if (e >= E) return;
  int s = src ? src[e] : e;
  int d = dst[e];
  const float4* xls = (const float4*)(xl + (long)s * C);
  const float4* xrd = (const float4*)(xr + (long)d * xr_stride);
  const float4* at4 = (const float4*)att;
  const float4* we0 = (const float4*)We;
  const float4* we1 = (const float4*)(We + C);
  float e0 = 0.f, e1 = 0.f;
  if (ea) {
    e0 = ea[(long)e * edge_dim];
    if (edge_dim > 1) e1 = ea[(long)e * edge_dim + 1];
  }
  float acc = 0.f;
  int C4 = C >> 2;
  for (int c = 0; c < C4; ++c) {
    float4 l = xls[c];
    float4 rr = xrd[c];
    float4 av = at4[c];
    float mx = l.x + rr.x, my = l.y + rr.y, mz = l.z + rr.z, mw = l.w + rr.w;
    if (ea) {
      float4 w0 = we0[c];
      mx += e0 * w0.x; my += e0 * w0.y; mz += e0 * w0.z; mw += e0 * w0.w;
      if (edge_dim > 1) {
        float4 w1 = we1[c];
        mx += e1 * w1.x; my += e1 * w1.y; mz += e1 * w1.z; mw += e1 * w1.w;
      }
    }
    acc += (mx > 0.f ? mx : NEG_GAT * mx) * av.x;
    acc += (my > 0.f ? my : NEG_GAT * my) * av.y;
    acc += (mz > 0.f ? mz : NEG_GAT * mz) * av.z;
    acc += (mw > 0.f ? mw : NEG_GAT * mw) * av.w;
  }
  score[e] = acc;
  atomicMax(&maxbuf[d], f2ord(acc));
}

__global__ void edge_softmax_kernel(const float* __restrict__ score,
                                    const int* __restrict__ dst,
                                    const unsigned int* __restrict__ maxbuf,
                                    float* __restrict__ exb,
                                    float* __restrict__ sumbuf, int E) {
  int e = blockIdx.x * blockDim.x + threadIdx.x;
  if (e >= E) return;
  int d = dst[e];
  float ex = __expf(score[e] - ord2f(maxbuf[d]));
  exb[e] = ex;
  atomicAdd(&sumbuf[d], ex);
}

__global__ void edge_scatter_kernel(const float* __restrict__ xl,
                                    const int* __restrict__ src,
                                    const int* __restrict__ dst,
                                    const float* __restrict__ exb,
                                    const float* __restrict__ sumbuf,
                                    float* __restrict__ out, int E, int C) {
  long idx = blockIdx.x * (long)blockDim.x + threadIdx.x;
  int cg = C >> 2;
  long total = (long)E * cg;
  if (idx >= total) return;
  int e = (int)(idx / cg);
  int c = (int)(idx % cg) * 4;
  int s = src ? src[e] : e;
  int d = dst[e];
  float alpha = exb[e] / sumbuf[d];
  float4 l = *(const float4*)(xl + (long)s * C + c);
  float* o = out + (long)d * C + c;
  atomicAdd(o + 0, alpha * l.x);
  atomicAdd(o + 1, alpha * l.y);
  atomicAdd(o + 2, alpha * l.z);
  atomicAdd(o + 3, alpha * l.w);
}

// h = leaky(h + out + bias_a (+ bias_b), 0.01), float4-wide
__global__ void residual_update_kernel(float* __restrict__ h,
                                       const float* __restrict__ out,
                                       const float* __restrict__ bias_a,
                                       const float* __restrict__ bias_b,
                                       long NC4, int C) {
  long idx = blockIdx.x * (long)blockDim.x + threadIdx.x;
  if (idx >= NC4) return;
  long base = idx * 4;
  int c = (int)(base % C);
  float4 hv = *(float4*)(h + base);
  float4 ov = *(const float4*)(out + base);
  float4 ba = *(const float4*)(bias_a + c);
  float4 bb = bias_b ? *(const float4*)(bias_b + c) : make_float4(0.f, 0.f, 0.f, 0.f);
  float vx = hv.x + ov.x + ba.x + bb.x;
  float vy = hv.y + ov.y + ba.y + bb.y;
  float vz = hv.z + ov.z + ba.z + bb.z;
  float vw = hv.w + ov.w + ba.w + bb.w;
  hv.x = vx > 0.f ? vx : NEG_LR * vx;
  hv.y = vy > 0.f ? vy : NEG_LR * vy;
  hv.z = vz > 0.f ? vz : NEG_LR * vz;
  hv.w = vw > 0.f ? vw : NEG_LR * vw;
  *(float4*)(h + base) = hv;
}

// dst = out + bias (in-place safe), float4-wide
__global__ void add_bias_kernel(float* __restrict__ dstp,
                                const float* __restrict__ out,
                                const float* __restrict__ bias,
                                long NC4, int C) {
  long idx = blockIdx.x * (long)blockDim.x + threadIdx.x;
  if (idx >= NC4) return;
  long base = idx * 4;
  int c = (int)(base % C);
  float4 ov = *(const float4*)(out + base);
  float4 bv = *(const float4*)(bias + c);
  ov.x += bv.x; ov.y += bv.y; ov.z += bv.z; ov.w += bv.w;
  *(float4*)(dstp + base) = ov;
}

// xr_row[c] = tok @ Wr[:,c] + br[c]   (B identical rows -> compute once)
__global__ void token_xr_kernel(const float* __restrict__ tok,
                                const float* __restrict__ Wr,
                                const float* __restrict__ br,
                                float* __restrict__ xr_row, int Ci, int Co) {
  int c = blockIdx.x * blockDim.x + threadIdx.x;
  if (c >= Co) return;
  float acc = br[c];
  for (int g = 0; g < Ci; ++g) acc += tok[g] * Wr[(long)g * Co + c];
  xr_row[c] = acc;
}

// cat[b, :] = [a[b,:], bm[b,:], cm[b,:]]
__global__ void concat3_kernel(const float* __restrict__ a,
                               const float* __restrict__ bm,
                               const float* __restrict__ cm,
                               float* __restrict__ cat, int Brows, int Gc) {
  int idx = blockIdx.x * blockDim.x + threadIdx.x;
  int tot = Brows * 3 * Gc;
  if (idx >= tot) return;
  int row = idx / (3 * Gc);
  int k = idx % (3 * Gc);
  float v;
  if (k < Gc)            v = a[row * Gc + k];
  else if (k < 2 * Gc)   v = bm[row * Gc + k - Gc];
  else                   v = cm[row * Gc + k - 2 * Gc];
  cat[idx] = v;
}

// ---------------------------------------------------------------------------
// Host orchestration
// ---------------------------------------------------------------------------
static inline void launch_gemm(const float* A, const float* W, const float* bias,
                               float* C, int N, int K, int CO, int act,
                               hipStream_t s) {
  dim3 grid((N + 63) / 64, CO / 64);
  gemm_wmma_f32<<<grid, 128, 0, s>>>(A, W, bias, C, N, K, CO, act);
}

static void run_gatv2(hipStream_t s,
                      const float* h_src, int Ns, int Cin_s,
                      const float* h_dst, int Nd, int Cin_d,
                      const int* src, const int* dst, int E,
                      const float* ea, const float* We, int edge_dim,
                      const float* Wl, const float* bl,
                      const float* Wr, const float* br,
                      const float* att, int Cout,
                      float* XL, float* XR, float* score, float* exb,
                      unsigned int* maxbuf, float* sumbuf,
                      float* out /* caller-zeroed accumulator [Nd,Cout] */) {
  launch_gemm(h_src, Wl, bl, XL, Ns, Cin_s, Cout, 0, s);
  launch_gemm(h_dst, Wr, br, XR, Nd, Cin_d, Cout, 0, s);
  hipMemsetAsync(maxbuf, 0, (size_t)Nd * 4, s);  // f2ord keys: 0 < any real key
  hipMemsetAsync(sumbuf, 0, (size_t)Nd * 4, s);
  int nb = (E + 255) / 256;
  edge_score_kernel<<<nb, 256, 0, s>>>(XL, XR, Cout, src, dst, ea, We, edge_dim,
                                       att, score, maxbuf, E, Cout);
  edge_softmax_kernel<<<nb, 256, 0, s>>>(score, dst, maxbuf, exb, sumbuf, E);
  long total = (long)E * (Cout >> 2);
  int nb2 = (int)((total + 255) / 256);
  edge_scatter_kernel<<<nb2, 256, 0, s>>>(XL, src, dst, exb, sumbuf, out, E, Cout);
}

extern "C" void kernel_launch(void* const* d_in, const int* in_sizes, int n_in,
                              void* d_out, int out_size, void* d_ws, size_t ws_size,
                              hipStream_t stream) {
  // ---- raw inputs (setup_inputs dict order) -------------------------------
  const float* x_op   = (const float*)d_in[0];
  const float* x_m    = (const float*)d_in[1];
  const float* x_a    = (const float*)d_in[2];
  const int*   ei_pr  = (const int*)d_in[3];   // [2, E_PP]
  const int*   ei_su  = (const int*)d_in[4];
  const int*   ei_om  = (const int*)d_in[5];
  const float* ea_om  = (const float*)d_in[6]; // [E_OM, 2]
  const int*   ei_ma  = (const int*)d_in[7];
  const float* ea_ma  = (const float*)d_in[8]; // [E_MA, 1]
  const int*   b_op   = (const int*)d_in[9];
  const int*   b_m    = (const int*)d_in[10];
  const int*   b_a    = (const int*)d_in[11];

  // ---- params: jax pytree leaves, dict keys sorted alphabetically ---------
  // layers[l]: ma{We,Wl,Wr,att,bias,bl,br} om{...7} pred{Wl,Wr,att,bias,bl,br} succ{...6}
  auto P = [&](int i) { return (const float*)d_in[i]; };
  const int LBASE = 12;                 // 26 leaves per layer
  const int MBASE = LBASE + NL * 26;    // 90: g_a(6) g_m(6) g_op(6) mix1(2) mix2(2) tok_a tok_m tok_op
  const int PBASE = MBASE + 25;         // 115: agv{W,b} machine{W,b} op{W,b}
  (void)n_in; (void)in_sizes; (void)out_size;

  // ---- d_out regions (h_* updated in place) -------------------------------
  float* out_f = (float*)d_out;
  float* h_op  = out_f;                       // [N_OP, 64]
  float* h_m   = h_op + (long)N_OP * NH;      // [N_M, 64]
  float* h_a   = h_m + (long)N_M * NH;        // [N_A, 64]
  float* g_op  = h_a + (long)N_A * NH;        // [64, 128]
  float* g_m   = g_op + NB * NG;
  float* g_a   = g_m + NB * NG;
  float* graph = g_a + NB * NG;               // [64, 256]

  // ---- workspace layout (~115 MB) -----------------------------------------
  float* ws = (float*)d_ws;
  float* XL     = ws;                   ws += (long)N_OP * NG;   // 12.8M
  float* XR     = ws;                   ws += (long)N_OP * NH;   // 6.4M
  float* OUT_OP = ws;                   ws += (long)N_OP * NH;   // 6.4M
  float* OUT_M  = ws;                   ws += (long)N_M * NH;
  float* OUT_A  = ws;                   ws += (long)N_A * NH;
  float* SCORE  = ws;                   ws += E_PP;
  float* EXB    = ws;                   ws += E_PP;
  float* SUMF   = ws;                   ws += N_OP;
  unsigned int* MAXU = (unsigned int*)ws; ws += N_OP;
  float* XRROW  = ws;                   ws += NG;
  float* CATB   = ws;                   ws += NB * 3 * NG;
  float* HID    = ws;                   ws += NB * NGG;
  (void)ws_size;

  // ---- 1. projections (WMMA GEMMs, K = 12 / 8) ----------------------------
  launch_gemm(x_op, P(PBASE + 4), P(PBASE + 5), h_op, N_OP, 12, NH, 0, stream);
  launch_gemm(x_m,  P(PBASE + 2), P(PBASE + 3), h_m,  N_M,  8,  NH, 0, stream);
  launch_gemm(x_a,  P(PBASE + 0), P(PBASE + 1), h_a,  N_A,  8,  NH, 0, stream);

  // ---- 2. three GATv2 layers ----------------------------------------------
  for (int l = 0; l < NL; ++l) {
    int L = LBASE + l * 26;
    const float *maWe = P(L+0),  *maWl = P(L+1),  *maWr = P(L+2),  *maAtt = P(L+3),
                *maB  = P(L+4),  *maBl = P(L+5),  *maBr = P(L+6);
    const float *omWe = P(L+7),  *omWl = P(L+8),  *omWr = P(L+9),  *omAtt = P(L+10),
                *omB  = P(L+11), *omBl = P(L+12), *omBr = P(L+13);
    const float *prWl = P(L+14), *prWr = P(L+15), *prAtt = P(L+16),
                *prB  = P(L+17), *prBl = P(L+18), *prBr = P(L+19);
    const float *suWl = P(L+20), *suWr = P(L+21), *suAtt = P(L+22),
                *suB  = P(L+23), *suBl = P(L+24), *suBr = P(L+25);

    hipMemsetAsync(OUT_OP, 0, (size_t)N_OP * NH * 4, stream);
    hipMemsetAsync(OUT_M,  0, (size_t)N_M * NH * 4, stream);
    hipMemsetAsync(OUT_A,  0, (size_t)N_A * NH * 4, stream);

    // op->op pred + succ accumulate into OUT_OP (HeteroConv sum)
    run_gatv2(stream, h_op, N_OP, NH, h_op, N_OP, NH,
              ei_pr, ei_pr + E_PP, E_PP, nullptr, nullptr, 0,
              prWl, prBl, prWr, prBr, prAtt, NH,
              XL, XR, SCORE, EXB, MAXU, SUMF, OUT_OP);
    run_gatv2(stream, h_op, N_OP, NH, h_op, N_OP, NH,
              ei_su, ei_su + E_PP, E_PP, nullptr, nullptr, 0,
              suWl, suBl, suWr, suBr, suAtt, NH,
              XL, XR, SCORE, EXB, MAXU, SUMF, OUT_OP);
    // op->machine (edge_dim 2)
    run_gatv2(stream, h_op, N_OP, NH, h_m, N_M, NH,
              ei_om, ei_om + E_OM, E_OM, ea_om, omWe, 2,
              omWl, omBl, omWr, omBr, omAtt, NH,
              XL, XR, SCORE, EXB, MAXU, SUMF, OUT_M);
    // machine->agv (edge_dim 1)
    run_gatv2(stream, h_m, N_M, NH, h_a, N_A, NH,
              ei_ma, ei_ma + E_MA, E_MA, ea_ma, maWe, 1,
              maWl, maBl, maWr, maBr, maAtt, NH,
              XL, XR, SCORE, EXB, MAXU, SUMF, OUT_A);

    // residual + leaky (after ALL relations read pre-update h)
    long q_op = (long)N_OP * NH / 4, q_m = (long)N_M * NH / 4, q_a = (long)N_A * NH / 4;
    residual_update_kernel<<<(int)((q_op + 255) / 256), 256, 0, stream>>>(
        h_op, OUT_OP, prB, suB, q_op, NH);
    residual_update_kernel<<<(int)((q_m + 255) / 256), 256, 0, stream>>>(
        h_m, OUT_M, omB, nullptr, q_m, NH);
    residual_update_kernel<<<(int)((q_a + 255) / 256), 256, 0, stream>>>(
        h_a, OUT_A, maB, nullptr, q_a, NH);
  }

  // ---- 3. graph-token readout (xr broadcast row, identity src) ------------
  struct Mix { const float* h; int N; const int* batch; int gbase; int tok; float* g; };
  Mix mixes[3] = {
      {h_op, N_OP, b_op, MBASE + 12, MBASE + 24, g_op},  // g_op, tok_op
      {h_m,  N_M,  b_m,  MBASE + 6,  MBASE + 23, g_m},   // g_m,  tok_m
      {h_a,  N_A,  b_a,  MBASE + 0,  MBASE + 22, g_a},   // g_a,  tok_a
  };
  for (int t = 0; t < 3; ++t) {
    const float *Wl = P(mixes[t].gbase + 0), *Wr = P(mixes[t].gbase + 1),
                *att = P(mixes[t].gbase + 2), *bias = P(mixes[t].gbase + 3),
                *bl = P(mixes[t].gbase + 4), *br = P(mixes[t].gbase + 5);
    const float* tok = P(mixes[t].tok);
    int Nt = mixes[t].N;
    launch_gemm(mixes[t].h, Wl, bl, XL, Nt, NH, NG, 0, stream);   // xl [Nt,128]
    token_xr_kernel<<<1, NG, 0, stream>>>(tok, Wr, br, XRROW, NG, NG);
    hipMemsetAsync(MAXU, 0, NB * 4, stream);
    hipMemsetAsync(SUMF, 0, NB * 4, stream);
    hipMemsetAsync(mixes[t].g, 0, (size_t)NB * NG * 4, stream);
    int nb = (Nt + 255) / 256;
    edge_score_kernel<<<nb, 256, 0, stream>>>(
        XL, XRROW, 0, nullptr, mixes[t].batch, nullptr, nullptr, 0,
        att, SCORE, MAXU, Nt, NG);
    edge_softmax_kernel<<<nb, 256, 0, stream>>>(
        SCORE, mixes[t].batch, MAXU, EXB, SUMF, Nt);
    long total = (long)Nt * (NG >> 2);
    edge_scatter_kernel<<<(int)((total + 255) / 256), 256, 0, stream>>>(
        XL, nullptr, mixes[t].batch, EXB, SUMF, mixes[t].g, Nt, NG);
    long q = (long)NB * NG / 4;
    add_bias_kernel<<<(int)((q + 255) / 256), 256, 0, stream>>>(
        mixes[t].g, mixes[t].g, bias, q, NG);
  }

  // ---- 4. head: cat -> mix1 (leaky) -> mix2 -------------------------------
  int tot = NB * 3 * NG;
  concat3_kernel<<<(tot + 255) / 256, 256, 0, stream>>>(g_op, g_m, g_a, CATB, NB, NG);
  launch_gemm(CATB, P(MBASE + 18), P(MBASE + 19), HID,   NB, 3 * NG, NGG, 1, stream);
  launch_gemm(HID,  P(MBASE + 20), P(MBASE + 21), graph, NB, NGG,   NGG, 0, stream);
}